// AttentionGuidedConv_83210696393639
// MI455X (gfx1250) — compile-verified
//
#include <hip/hip_runtime.h>

// Problem sizes (B==C==96 so the conv tap table is per-channel)
#define B_    96
#define L_    8192
#define C_    96
#define K_    31
#define A_    512
#define LOUT  8162            // L - K + 1

// Conv tiling: one workgroup = one batch b, one 256-wide t tile, all 96 channels.
// WMMA D tile is 16x16: t = t0 + 16*m + n. Toeplitz contraction length 48 -> 12 K=4 steps.
#define TILE_T 256
#define NT     32             // ceil(8162/256)
#define XROWS  288            // x rows needed: 16*15 + 47 + 1
#define ST     290            // LDS floats per channel row (even -> 8B-aligned b64 loads)
#define KERP   64             // zero-padded tap row: kp[c][j] = ker[c][j-16] for j-16 in [0,31)

typedef __attribute__((ext_vector_type(2))) float v2f;
typedef __attribute__((ext_vector_type(4))) float v4f;
typedef __attribute__((ext_vector_type(8))) float v8f;

// low 32 bits of a generic pointer derived from __shared__ == LDS byte offset
__device__ __forceinline__ unsigned lds_off(const void* p) {
  return (unsigned)(unsigned long long)p;
}

// ---------------------------------------------------------------------------
// Stage 1: write the zero-padded Toeplitz tap table kp[96][64] into d_ws:
//   kp[c][j] = attn[c,:] . proj_w[j-16,:] + proj_b[j-16]  for j-16 in [0,31), else 0
// ---------------------------------------------------------------------------
__global__ void kernels_gemm(const float* __restrict__ attw,
                             const float* __restrict__ pw,
                             const float* __restrict__ pb,
                             float* __restrict__ kp) {
  int id = blockIdx.x * blockDim.x + threadIdx.x;   // 96*64 threads
  if (id >= C_ * KERP) return;
  int c = id >> 6, j = id & 63;
  int k = j - 16;
  float out = 0.f;
  if (k >= 0 && k < K_) {
    const v4f* wa = (const v4f*)(attw + (size_t)c * A_);
    const v4f* wp = (const v4f*)(pw   + (size_t)k * A_);
    float acc = 0.f;
#pragma unroll 4
    for (int a = 0; a < A_ / 4; ++a) {
      v4f u = wa[a], w = wp[a];
      acc = fmaf(u.x, w.x, acc);
      acc = fmaf(u.y, w.y, acc);
      acc = fmaf(u.z, w.z, acc);
      acc = fmaf(u.w, w.w, acc);
    }
    out = acc + pb[k];
  }
  kp[id] = out;
}

// ---------------------------------------------------------------------------
// Stage 2: depthwise conv via per-channel Toeplitz WMMA (fp32, 16x16x4)
//   D[m,n] = sum_q A[m,q] * B[q,n],  A[m,q] = x[t0+16m+q, c],  B[q,n] = kp[c][q-n+16]
// LDS x tile: channel-major, XOR swizzle on t bits[3:2] by bits[7:6] ->
// stride-16 A fetches bank-conflict-free, k-pairs stay b64-contiguous.
// Staging uses gfx1250 async-to-LDS (ASYNCcnt), no VGPR round trip.
// ---------------------------------------------------------------------------
__launch_bounds__(256)
__global__ void dwconv_wmma(const float* __restrict__ x,
                            const float* __restrict__ kpg,
                            float* __restrict__ y) {
  __shared__ __align__(16) float xs[C_ * ST];    // 111,360 B
  __shared__ __align__(16) float kp[C_ * KERP];  //  24,576 B

  const int b  = blockIdx.x >> 5;
  const int t0 = (blockIdx.x & 31) * TILE_T;

  const int tid  = threadIdx.x;
  const int lane = tid & 31;
  const int wave = tid >> 5;

  const unsigned xs_base = lds_off(xs);
  const unsigned kp_base = lds_off(kp);

  // ---- tap table -> LDS: contiguous async b128 copy (1536 float4 = 6*256) ----
  for (int i = tid; i < (C_ * KERP) / 4; i += 256) {
    unsigned ldst = kp_base + 16u * (unsigned)i;
    unsigned goff = 16u * (unsigned)i;
    asm volatile("global_load_async_to_lds_b128 %0, %1, %2 offset:0"
                 :: "v"(ldst), "v"(goff), "s"(kpg) : "memory");
  }

  // ---- x tile -> LDS: transpose [t][c] -> [c][t'] swizzled, async b32 scatter ----
  const float* xb = x + (size_t)b * L_ * C_;
  const bool interior = (t0 + XROWS) <= L_;        // uniform per block
  for (int i = tid; i < XROWS * (C_ / 4); i += 256) {   // 6912 = 27*256
    int r  = i / (C_ / 4);
    int c4 = (i % (C_ / 4)) * 4;
    int rg = t0 + r;
    if (!interior && rg > L_ - 1) rg = L_ - 1;     // clamp: only feeds B==0 terms
    unsigned goff = ((unsigned)rg * C_ + (unsigned)c4) * 4u;
    int m = r >> 4, tl = r & 15;
    int off = 16 * m + (tl ^ (((m >> 2) & 3) << 2));
    unsigned l0 = xs_base + 4u * (unsigned)((c4 + 0) * ST + off);
    unsigned l1 = xs_base + 4u * (unsigned)((c4 + 1) * ST + off);
    unsigned l2 = xs_base + 4u * (unsigned)((c4 + 2) * ST + off);
    unsigned l3 = xs_base + 4u * (unsigned)((c4 + 3) * ST + off);
    asm volatile(
        "global_load_async_to_lds_b32 %0, %4, %5 offset:0\n\t"
        "global_load_async_to_lds_b32 %1, %4, %5 offset:4\n\t"
        "global_load_async_to_lds_b32 %2, %4, %5 offset:8\n\t"
        "global_load_async_to_lds_b32 %3, %4, %5 offset:12"
        :: "v"(l0), "v"(l1), "v"(l2), "v"(l3), "v"(goff), "s"(xb)
        : "memory");
  }

  asm volatile("s_wait_asynccnt 0x0" ::: "memory");
  __syncthreads();

  const int n    = lane & 15;
  const int half = lane >> 4;
  const int m    = n;                     // A-matrix row for this lane
  const bool full_tile = (t0 + TILE_T) <= LOUT;    // uniform per block

  // each wave owns 12 channels
  for (int cc = 0; cc < 12; ++cc) {
    const int c = wave * 12 + cc;

    // Toeplitz B for this channel: 12 steps x 2 regs (predication-free gather)
    float bv[24];
    const float* kpc = kp + c * KERP + (16 - n + 2 * half);
#pragma unroll
    for (int s = 0; s < 12; ++s) {
      bv[2 * s + 0] = kpc[4 * s + 0];
      bv[2 * s + 1] = kpc[4 * s + 1];
    }

    v8f acc = {0.f, 0.f, 0.f, 0.f, 0.f, 0.f, 0.f, 0.f};
    const float* xc = xs + c * ST;
#pragma unroll
    for (int s = 0; s < 12; ++s) {
      int q  = 4 * s + 2 * half;          // even, < 48
      int mp = m + (q >> 4);
      int ql = q & 15;
      int sw = ql ^ (((mp >> 2) & 3) << 2);
      v2f a = *(const v2f*)(xc + 16 * mp + sw);   // ds_load_b64, conflict-free
      v2f bb; bb.x = bv[2 * s]; bb.y = bv[2 * s + 1];
      acc = __builtin_amdgcn_wmma_f32_16x16x4_f32(
          /*neg_a=*/false, a, /*neg_b=*/false, bb,
          /*c_mod=*/(short)0, acc, /*reuse_a=*/false, /*reuse_b=*/false);
    }

    // D: vgpr v holds y[t0 + 16*(v + 8*half) + n, c]
    float* yb = y + (size_t)b * LOUT * C_ + c;
    if (full_tile) {
#pragma unroll
      for (int v = 0; v < 8; ++v) {
        int t = t0 + 16 * (v + 8 * half) + n;
        yb[(size_t)t * C_] = acc[v];
      }
    } else {
#pragma unroll
      for (int v = 0; v < 8; ++v) {
        int t = t0 + 16 * (v + 8 * half) + n;
        if (t < LOUT) yb[(size_t)t * C_] = acc[v];
      }
    }
  }
}

// ---------------------------------------------------------------------------
extern "C" void kernel_launch(void* const* d_in, const int* in_sizes, int n_in,
                              void* d_out, int out_size, void* d_ws, size_t ws_size,
                              hipStream_t stream) {
  const float* x    = (const float*)d_in[0];
  const float* attw = (const float*)d_in[1];
  const float* pw   = (const float*)d_in[2];
  const float* pb   = (const float*)d_in[3];
  float* y   = (float*)d_out;
  float* kpw = (float*)d_ws;                 // 96*64 floats: zero-padded tap table

  kernels_gemm<<<(C_ * KERP + 255) / 256, 256, 0, stream>>>(attw, pw, pb, kpw);
  dwconv_wmma<<<B_ * NT, 256, 0, stream>>>(x, kpw, y);
}